// MambaDepthCell_67877663146033
// MI455X (gfx1250) — compile-verified
//
#include <hip/hip_runtime.h>
#include <math.h>

// ---------------- problem constants (from reference) ----------------
constexpr int kL   = 2;
constexpr int kT   = 1024;          // B*P
constexpr int kH   = 768;
constexpr int kDIN = 1536;          // E*H
constexpr int kNs  = 16;            // SSM state N
constexpr int kKc  = 4;             // conv kernel width
constexpr int kR   = 48;            // dt rank
constexpr int kXDB = kR + 2 * kNs;  // 80

typedef __attribute__((ext_vector_type(2))) float v2f;
typedef __attribute__((ext_vector_type(8))) float v8f;

__device__ __forceinline__ float siluf(float x) {
    return x / (1.0f + expf(-x));
}
__device__ __forceinline__ float softplusf(float x) {
    return (x > 20.0f) ? x : log1pf(expf(x));
}

__device__ __forceinline__ v8f wmma4(v2f a, v2f b, v8f c) {
    return __builtin_amdgcn_wmma_f32_16x16x4_f32(
        false, a, false, b, (short)0, c, false, false);
}

// ---------------- fp32 WMMA GEMM: C(MxN) = A(MxK) * W(NxK)^T ----------------
// MODE 0: plain   MODE 1: += bias[n], softplus   MODE 2: += C (residual, in place)
// Wave tile: 32(M) x 32(N) = 2x2 accumulators -> 1.0 b64 loads per WMMA.
// Block: 8 waves as 4(M) x 2(N) -> block tile 128 x 64.
template <int MODE>
__global__ __launch_bounds__(256) void gemm_wmma_f32(
    const float* __restrict__ A, int lda,
    const float* __restrict__ W, int ldw,
    float* __restrict__ C, int ldc,
    const float* __restrict__ bias,
    int M, int N, int K)
{
    const int wave = threadIdx.x >> 5;
    const int lane = threadIdx.x & 31;
    const int m0 = (blockIdx.y * 4 + (wave & 3)) * 32;
    const int n0 = (blockIdx.x * 2 + (wave >> 2)) * 32;
    if (m0 >= M || n0 >= N) return;        // wave-uniform exit: EXEC all-ones for WMMA
    const bool n1ok = (n0 + 16) < N;       // wave-uniform subtile mask (N=80 case)

    const int half = lane >> 4;   // 0: K pair {0,1}, 1: K pair {2,3}
    const int row  = lane & 15;   // M row (A) / N col (B)
    const float* Ap0 = A + (size_t)(m0 + row) * lda + (half << 1);
    const float* Ap1 = Ap0 + (size_t)16 * lda;
    const float* Wp0 = W + (size_t)(n0 + row) * ldw + (half << 1);

    v8f acc00 = {}, acc01 = {}, acc10 = {}, acc11 = {};

    if (n1ok) {
        const float* Wp1 = Wp0 + (size_t)16 * ldw;
        for (int k = 0; k < K; k += 16) {
#pragma unroll
            for (int kk = 0; kk < 16; kk += 4) {
                v2f a0 = *(const v2f*)(Ap0 + k + kk);
                v2f a1 = *(const v2f*)(Ap1 + k + kk);
                v2f b0 = *(const v2f*)(Wp0 + k + kk);
                v2f b1 = *(const v2f*)(Wp1 + k + kk);
                acc00 = wmma4(a0, b0, acc00);
                acc10 = wmma4(a1, b0, acc10);
                acc01 = wmma4(a0, b1, acc01);
                acc11 = wmma4(a1, b1, acc11);
            }
        }
    } else {
        for (int k = 0; k < K; k += 16) {
#pragma unroll
            for (int kk = 0; kk < 16; kk += 4) {
                v2f a0 = *(const v2f*)(Ap0 + k + kk);
                v2f a1 = *(const v2f*)(Ap1 + k + kk);
                v2f b0 = *(const v2f*)(Wp0 + k + kk);
                acc00 = wmma4(a0, b0, acc00);
                acc10 = wmma4(a1, b0, acc10);
            }
        }
    }

    // ---------------- epilogue: up to 4 16x16 tiles ----------------
    const int nA = n0 + row;
    const int nB = n0 + 16 + row;
#pragma unroll
    for (int r = 0; r < 8; ++r) {
        const int mA = m0 + r + (half << 3);      // VGPR r -> rows r / r+8
        const int mB = mA + 16;
        float v00 = acc00[r], v10 = acc10[r];
        if constexpr (MODE == 1) {
            v00 = softplusf(v00 + bias[nA]);
            v10 = softplusf(v10 + bias[nA]);
        }
        if constexpr (MODE == 2) {
            v00 += C[(size_t)mA * ldc + nA];
            v10 += C[(size_t)mB * ldc + nA];
        }
        C[(size_t)mA * ldc + nA] = v00;
        C[(size_t)mB * ldc + nA] = v10;
        if (n1ok) {
            float v01 = acc01[r], v11 = acc11[r];
            if constexpr (MODE == 1) {
                v01 = softplusf(v01 + bias[nB]);
                v11 = softplusf(v11 + bias[nB]);
            }
            if constexpr (MODE == 2) {
                v01 += C[(size_t)mA * ldc + nB];
                v11 += C[(size_t)mB * ldc + nB];
            }
            C[(size_t)mA * ldc + nB] = v01;
            C[(size_t)mB * ldc + nB] = v11;
        }
    }
}

// ---------------- rmsnorm: one block per row of (T, kH) ----------------
__global__ __launch_bounds__(256) void rmsnorm_kernel(
    const float* __restrict__ x, const float* __restrict__ w,
    float* __restrict__ out)
{
    const int t = blockIdx.x;
    const float* xr = x + (size_t)t * kH;
    float vals[3];
    float s = 0.0f;
#pragma unroll
    for (int i = 0; i < 3; ++i) {
        vals[i] = xr[threadIdx.x + i * 256];
        s += vals[i] * vals[i];
    }
#pragma unroll
    for (int off = 16; off >= 1; off >>= 1) s += __shfl_down(s, off, 32);
    __shared__ float red[8];
    if ((threadIdx.x & 31) == 0) red[threadIdx.x >> 5] = s;
    __syncthreads();
    float tot = 0.0f;
#pragma unroll
    for (int i = 0; i < 8; ++i) tot += red[i];
    const float scale = rsqrtf(tot / (float)kH + 1e-5f);
#pragma unroll
    for (int i = 0; i < 3; ++i) {
        const int c = threadIdx.x + i * 256;
        out[(size_t)t * kH + c] = vals[i] * scale * w[c];
    }
}

// ---------------- conv state shift + depthwise conv + SiLU ----------------
__global__ __launch_bounds__(256) void conv_kernel(
    const float* __restrict__ conv_state_in,  // (T, DIN, 4) layer slice
    const float* __restrict__ proj,           // (T, 2*DIN): [:,:DIN]=hs_in
    const float* __restrict__ conv_w,         // (DIN, 4)
    const float* __restrict__ conv_b,         // (DIN)
    float* __restrict__ conv_state_out,       // (T, DIN, 4)
    float* __restrict__ hs_out)               // (T, DIN)
{
    const int idx = blockIdx.x * 256 + threadIdx.x;
    if (idx >= kT * kDIN) return;
    const int t = idx / kDIN;
    const int d = idx - t * kDIN;

    float4 cs = *(const float4*)(conv_state_in + (size_t)idx * kKc);
    const float hin = proj[(size_t)t * (2 * kDIN) + d];
    float4 ncs;
    ncs.x = cs.y; ncs.y = cs.z; ncs.z = cs.w; ncs.w = hin;
    *(float4*)(conv_state_out + (size_t)idx * kKc) = ncs;

    const float4 w = *(const float4*)(conv_w + (size_t)d * kKc);
    const float acc = ncs.x * w.x + ncs.y * w.y + ncs.z * w.z + ncs.w * w.w + conv_b[d];
    hs_out[idx] = siluf(acc);
}

// ---------------- SSM state update + C-reduction + gate ----------------
__global__ __launch_bounds__(256) void ssm_kernel(
    const float* __restrict__ ssm_in,   // (T, DIN, 16) layer slice
    const float* __restrict__ dt,       // (T, DIN)  (softplus applied)
    const float* __restrict__ hs,       // (T, DIN)
    const float* __restrict__ xdb,      // (T, 80): [:,48:64]=B, [:,64:80]=C
    const float* __restrict__ A_log,    // (DIN, 16) layer slice
    const float* __restrict__ Dp,       // (DIN)
    const float* __restrict__ proj,     // (T, 2*DIN): [:,DIN:]=gate
    float* __restrict__ ssm_out,        // (T, DIN, 16)
    float* __restrict__ y_out)          // (T, DIN)
{
    const int idx = blockIdx.x * 256 + threadIdx.x;
    if (idx >= kT * kDIN) return;
    const int t = idx / kDIN;
    const int d = idx - t * kDIN;

    const float dtv = dt[idx];
    const float hv  = hs[idx];
    const float* Bm = xdb + (size_t)t * kXDB + kR;
    const float* Cm = Bm + kNs;
    const float* Al = A_log + (size_t)d * kNs;
    const float* si = ssm_in + (size_t)idx * kNs;
    float* so       = ssm_out + (size_t)idx * kNs;

    float y = 0.0f;
#pragma unroll
    for (int n = 0; n < kNs; ++n) {
        const float dA  = expf(-expf(Al[n]) * dtv);
        const float ssv = si[n] * dA + dtv * Bm[n] * hv;
        so[n] = ssv;
        y += ssv * Cm[n];
    }
    y += Dp[d] * hv;
    const float g = proj[(size_t)t * (2 * kDIN) + kDIN + d];
    y_out[idx] = y * siluf(g);
}

// ---------------- host launcher ----------------
extern "C" void kernel_launch(void* const* d_in, const int* in_sizes, int n_in,
                              void* d_out, int out_size, void* d_ws, size_t ws_size,
                              hipStream_t stream) {
    const float* u           = (const float*)d_in[0];
    const float* conv_states = (const float*)d_in[1];
    const float* ssm_states  = (const float*)d_in[2];
    const float* in_proj_w   = (const float*)d_in[3];
    const float* conv_w      = (const float*)d_in[4];
    const float* conv_b      = (const float*)d_in[5];
    const float* x_proj_w    = (const float*)d_in[6];
    const float* dt_proj_w   = (const float*)d_in[7];
    const float* dt_proj_b   = (const float*)d_in[8];
    const float* A_log       = (const float*)d_in[9];
    const float* Dm          = (const float*)d_in[10];
    const float* out_proj_w  = (const float*)d_in[11];
    const float* norm_w      = (const float*)d_in[12];
    const float* norm_f_w    = (const float*)d_in[13];

    float* out      = (float*)d_out;
    float* out_x    = out;                                    // (T, H)
    float* out_conv = out_x + (size_t)kT * kH;                // (L, T, DIN, 4)
    float* out_ssm  = out_conv + (size_t)kL * kT * kDIN * kKc;// (L, T, DIN, 16)

    float* ws    = (float*)d_ws;
    float* x_cur = ws;                                  // (T, H)
    float* hn    = x_cur + (size_t)kT * kH;             // (T, H)
    float* proj  = hn + (size_t)kT * kH;                // (T, 2*DIN)
    float* hsbuf = proj + (size_t)kT * 2 * kDIN;        // (T, DIN)
    float* xdb   = hsbuf + (size_t)kT * kDIN;           // (T, 80)
    float* dtbuf = xdb + (size_t)kT * kXDB;             // (T, DIN)
    float* ybuf  = dtbuf + (size_t)kT * kDIN;           // (T, DIN)

    // residual stream starts as u
    hipMemcpyAsync(x_cur, u, sizeof(float) * (size_t)kT * kH,
                   hipMemcpyDeviceToDevice, stream);

    const int EW = kT * kDIN;
    const dim3 blk(256);
    const int mgrid = kT / 128;   // 8

    for (int l = 0; l < kL; ++l) {
        // hn = rmsnorm(x, norm_w[l])
        rmsnorm_kernel<<<kT, blk, 0, stream>>>(x_cur, norm_w + (size_t)l * kH, hn);

        // proj = hn @ in_proj_w[l]^T : M=T, N=2*DIN, K=H
        gemm_wmma_f32<0><<<dim3((2 * kDIN) / 64, mgrid), blk, 0, stream>>>(
            hn, kH, in_proj_w + (size_t)l * 2 * kDIN * kH, kH,
            proj, 2 * kDIN, nullptr, kT, 2 * kDIN, kH);

        // conv state shift + SiLU, writes new_conv
        conv_kernel<<<(EW + 255) / 256, blk, 0, stream>>>(
            conv_states + (size_t)l * kT * kDIN * kKc, proj,
            conv_w + (size_t)l * kDIN * kKc, conv_b + (size_t)l * kDIN,
            out_conv + (size_t)l * kT * kDIN * kKc, hsbuf);

        // xdb = hs @ x_proj_w[l]^T : M=T, N=80, K=DIN
        gemm_wmma_f32<0><<<dim3((kXDB + 63) / 64, mgrid), blk, 0, stream>>>(
            hsbuf, kDIN, x_proj_w + (size_t)l * kXDB * kDIN, kDIN,
            xdb, kXDB, nullptr, kT, kXDB, kDIN);

        // dt = softplus(xdb[:, :48] @ dt_proj_w[l]^T + b) : M=T, N=DIN, K=48
        gemm_wmma_f32<1><<<dim3(kDIN / 64, mgrid), blk, 0, stream>>>(
            xdb, kXDB, dt_proj_w + (size_t)l * kDIN * kR, kR,
            dtbuf, kDIN, dt_proj_b + (size_t)l * kDIN, kT, kDIN, kR);

        // SSM update, writes new_ssm and gated y
        ssm_kernel<<<(EW + 255) / 256, blk, 0, stream>>>(
            ssm_states + (size_t)l * kT * kDIN * kNs, dtbuf, hsbuf, xdb,
            A_log + (size_t)l * kDIN * kNs, Dm + (size_t)l * kDIN, proj,
            out_ssm + (size_t)l * kT * kDIN * kNs, ybuf);

        // x = x + y @ out_proj_w[l]^T : M=T, N=H, K=DIN (residual in place)
        gemm_wmma_f32<2><<<dim3(kH / 64, mgrid), blk, 0, stream>>>(
            ybuf, kDIN, out_proj_w + (size_t)l * kH * kDIN, kDIN,
            x_cur, kH, nullptr, kT, kH, kDIN);
    }

    // final rmsnorm -> output x
    rmsnorm_kernel<<<kT, blk, 0, stream>>>(x_cur, norm_f_w, out_x);
}